// YOLOv9Loss_4398046511284
// MI455X (gfx1250) — compile-verified
//
#include <hip/hip_runtime.h>
#include <hip/hip_bf16.h>
#include <cstdint>
#include <cfloat>

// =============================================================================
// YOLOv9 box-matcher + loss for MI455X (gfx1250, wave32).
// Roofline: ~1 GFLOP of CIoU/transcendental math over B*T*A=17.2M pairs and
// ~400MB of HBM traffic -> bandwidth/VALU bound, fp32 throughout.
// CDNA5 paths used: GLOBAL_LOAD_ASYNC_TO_LDS_B32 + s_wait_asynccnt (top-k row
// staging), global_prefetch_b8 (streaming loops), V_WMMA_F32_16X16X4_F32
// (ones-matrix wave32 partial-sum reduction). All behind __has_builtin guards.
// Workspace: tm buffer B*T*A floats (~69MB) + small per-(b,t)/(b,a) arrays.
// =============================================================================

#define EPSF      1e-7f
#define INV_PI2_4 0.405284734569f /* 4/pi^2 */
#define TOPK_N    10
#define A_LDS     8400
#define T_CAP     64
#define BLK       256

typedef __attribute__((ext_vector_type(2))) float v2f;
typedef __attribute__((ext_vector_type(8))) float v8f;

// ---------------- CDNA5 async global->LDS path -------------------------------
#if defined(__gfx1250__) && __has_builtin(__builtin_amdgcn_global_load_async_to_lds_b32)
#define HAVE_ASYNC_LDS 1
typedef __attribute__((address_space(1))) int as1_int;  // HIP prints AS1 as __device__
typedef __attribute__((address_space(3))) int as3_int;  // LDS (AS3), 32-bit pointers
__device__ __forceinline__ void async_g2l_b32(const float* g, float* l) {
  __builtin_amdgcn_global_load_async_to_lds_b32(
      (as1_int*)(uintptr_t)g,
      (as3_int*)(uint32_t)(uintptr_t)l, 0, 0);
}
__device__ __forceinline__ void wait_async0() {
#if __has_builtin(__builtin_amdgcn_s_wait_asynccnt)
  __builtin_amdgcn_s_wait_asynccnt(0);
#else
  asm volatile("s_wait_asynccnt 0" ::: "memory");
#endif
}
#endif

// ---------------- wave32 sum (WMMA-assisted when available) ------------------
// A = per-lane value in K=0 slot of a 16x4 f32 A-matrix, B = all-ones 4x16.
// C[m][n] = rowsum(m) = p(m) + p(m+16); per-lane sum of 8 C regs gives the
// half-wave total; one shfl_xor(16) completes the 32-lane sum.
__device__ __forceinline__ float waveSum32(float p) {
#if defined(__gfx1250__) && __has_builtin(__builtin_amdgcn_wmma_f32_16x16x4_f32)
  v2f a; a.x = p;    a.y = 0.0f;
  v2f b; b.x = 1.0f; b.y = 1.0f;
  v8f c = {};
  c = __builtin_amdgcn_wmma_f32_16x16x4_f32(false, a, false, b, (short)0, c,
                                            false, false);
  float s = c[0] + c[1] + c[2] + c[3] + c[4] + c[5] + c[6] + c[7];
  s += __shfl_xor(s, 16, 32);
  return s;
#else
  for (int o = 16; o > 0; o >>= 1) p += __shfl_xor(p, o, 32);
  return p;
#endif
}

__device__ __forceinline__ float blockReduceSum(float v) {
  __shared__ float sp[BLK / 32];
  float w = waveSum32(v);
  int lane = threadIdx.x & 31, wv = threadIdx.x >> 5;
  __syncthreads();               // protect sp across consecutive calls
  if (lane == 0) sp[wv] = w;
  __syncthreads();
  float r = 0.0f;
  if (wv == 0) {
    int nw = (blockDim.x + 31) >> 5;
    r = (lane < nw) ? sp[lane] : 0.0f;
    for (int o = 4; o > 0; o >>= 1) r += __shfl_xor(r, o, 32);
  }
  return r;  // valid on wave 0 (thread 0 uses it)
}

// ---------------- CIoU (symmetric in the two boxes; tat = atan(w1/(h1+eps))) -
__device__ __forceinline__ float ciou_eval(float tx1, float ty1, float tx2, float ty2,
                                           float tat,
                                           float px1, float py1, float px2, float py2) {
  float xi1 = fmaxf(tx1, px1), yi1 = fmaxf(ty1, py1);
  float xi2 = fminf(tx2, px2), yi2 = fminf(ty2, py2);
  float inter = fmaxf(xi2 - xi1, 0.0f) * fmaxf(yi2 - yi1, 0.0f);
  float w1 = tx2 - tx1, h1 = ty2 - ty1;
  float w2 = px2 - px1, h2 = py2 - py1;
  float uni = w1 * h1 + w2 * h2 - inter;
  float iou = inter / (uni + EPSF);
  float dx = (tx2 + tx1 - px2 - px1) * 0.5f;
  float dy = (ty2 + ty1 - py2 - py1) * 0.5f;
  float cent = dx * dx + dy * dy;
  float cw = fmaxf(tx2, px2) - fminf(tx1, px1);
  float ch = fmaxf(ty2, py2) - fminf(ty1, py1);
  float diag = cw * cw + ch * ch + EPSF;
  float diou = iou - cent / diag;
  float d = tat - atanf(w2 / (h2 + EPSF));
  float vv = INV_PI2_4 * d * d;
  float alpha = vv / (vv - iou + 1.0f + EPSF);
  return diou - alpha * vv;
}
__device__ __forceinline__ float clip01(float x) { return fminf(fmaxf(x, 0.0f), 1.0f); }

// ---------------- k0: zero accumulators --------------------------------------
__global__ void k_init(float* maxtm, float* maxiou, float* accum, int nbt) {
  int i = blockIdx.x * blockDim.x + threadIdx.x;
  if (i < nbt) { maxtm[i] = 0.0f; maxiou[i] = 0.0f; }
  if (i < 8) accum[i] = 0.0f;
}

// ---------------- k1: target_matrix[b,t,a] -----------------------------------
__global__ void k1_target_matrix(const float* __restrict__ target,
                                 const float* __restrict__ pcls,
                                 const float* __restrict__ pbox,
                                 const float* __restrict__ anchors,
                                 float* __restrict__ tm,
                                 int T, int A, int C) {
  int bt = blockIdx.x;
  int b = bt / T;
  const float* tr = target + (size_t)bt * 5;
  int tc = (int)tr[0]; tc = tc < 0 ? 0 : (tc >= C ? C - 1 : tc);
  float tx1 = tr[1], ty1 = tr[2], tx2 = tr[3], ty2 = tr[4];
  float tat = atanf((tx2 - tx1) / (ty2 - ty1 + EPSF));
  const float4* pb = (const float4*)(pbox + (size_t)b * A * 4);
  const float* pc = pcls + (size_t)b * A * C + tc;
  const float2* an = (const float2*)anchors;
  float* out = tm + (size_t)bt * A;
  for (int a = threadIdx.x; a < A; a += blockDim.x) {
    __builtin_prefetch(&pb[a + BLK], 0, 0);   // global_prefetch_b8
    float2 xy = an[a];
    float val = 0.0f;
    if (tx1 < xy.x && xy.x < tx2 && ty1 < xy.y && xy.y < ty2) {
      float4 p = pb[a];
      float iou = clip01(ciou_eval(tx1, ty1, tx2, ty2, tat, p.x, p.y, p.z, p.w));
      float x = pc[(size_t)a * C];
      float sg = 1.0f / (1.0f + expf(-x));
      float i2 = iou * iou;
      val = i2 * i2 * i2 * sqrtf(sg);           // iou^6 * cls^0.5
    }
    out[a] = val;
  }
}

// ---------------- k2: per-(b,t) 10th-largest threshold (lex tie-break) -------
__global__ void k2_topk(const float* __restrict__ tm, float* __restrict__ thr, int A) {
  __shared__ float srow[A_LDS];
  __shared__ float sv[BLK];
  __shared__ int si[BLK];
  int bt = blockIdx.x;
  int tid = threadIdx.x;
  const float* row = tm + (size_t)bt * A;
#ifdef HAVE_ASYNC_LDS
  for (int i = tid; i < A; i += BLK) async_g2l_b32(row + i, &srow[i]);
  wait_async0();
#else
  for (int i = tid; i < A; i += BLK) srow[i] = row[i];
#endif
  __syncthreads();
  float prevV = FLT_MAX;
  int prevI = -1;
  for (int k = 0; k < TOPK_N; ++k) {
    float bv = -1.0f; int bi = 0x7fffffff;
    for (int i = tid; i < A; i += BLK) {
      float v = srow[i];
      bool elig = (v < prevV) || (v == prevV && i > prevI);
      if (elig && (v > bv || (v == bv && i < bi))) { bv = v; bi = i; }
    }
    sv[tid] = bv; si[tid] = bi;
    __syncthreads();
    for (int s = BLK / 2; s > 0; s >>= 1) {
      if (tid < s) {
        float ov = sv[tid + s]; int oi = si[tid + s];
        if (ov > sv[tid] || (ov == sv[tid] && oi < si[tid])) { sv[tid] = ov; si[tid] = oi; }
      }
      __syncthreads();
    }
    prevV = sv[0]; prevI = si[0];
    __syncthreads();
  }
  if (tid == 0) thr[bt] = fmaxf(prevV, 0.0f);
}

// ---------------- k3: dedup + unique index + per-(b,t) masked maxima ---------
__global__ void k3_dedup(const float* __restrict__ target,
                         const float* __restrict__ pbox,
                         const float* __restrict__ tm,
                         const float* __restrict__ thr,
                         float* __restrict__ maxtm,
                         float* __restrict__ maxiou,
                         int* __restrict__ uidx,
                         float* __restrict__ tmstar,
                         int T, int A) {
  __shared__ float sx1[T_CAP], sy1[T_CAP], sx2[T_CAP], sy2[T_CAP], sat[T_CAP], sth[T_CAP];
  int b = blockIdx.y;
  int tid = threadIdx.x;
  if (tid < T) {
    const float* tr = target + ((size_t)b * T + tid) * 5;
    sx1[tid] = tr[1]; sy1[tid] = tr[2]; sx2[tid] = tr[3]; sy2[tid] = tr[4];
    sat[tid] = atanf((tr[3] - tr[1]) / (tr[4] - tr[2] + EPSF));
    sth[tid] = thr[b * T + tid];
  }
  __syncthreads();
  int a = blockIdx.x * blockDim.x + tid;
  if (a >= A) return;
  const float* col = tm + (size_t)b * T * A + a;
  int count = 0, first = -1;
  for (int t = 0; t < T; ++t) {
    float v = col[(size_t)t * A];
    if (v > 0.0f && v >= sth[t]) { count++; if (first < 0) first = t; }
  }
  int tstar = 0;
  float mark = -1.0f;  // <0 => invalid anchor
  if (count > 0) {
    float4 p = ((const float4*)(pbox + (size_t)b * A * 4))[a];
    if (count == 1) {
      tstar = first;
    } else {  // duplicates -> argmax_t of clipped iou (first index on ties)
      float best = -1.0f; int bt_ = 0;
      for (int t = 0; t < T; ++t) {
        float iou = clip01(ciou_eval(sx1[t], sy1[t], sx2[t], sy2[t], sat[t],
                                     p.x, p.y, p.z, p.w));
        if (iou > best) { best = iou; bt_ = t; }
      }
      tstar = bt_;
    }
    float tms = col[(size_t)tstar * A];
    float iouv = clip01(ciou_eval(sx1[tstar], sy1[tstar], sx2[tstar], sy2[tstar],
                                  sat[tstar], p.x, p.y, p.z, p.w));
    // values are >=0 so uint bit-pattern atomicMax == float max
    atomicMax((unsigned int*)&maxtm[b * T + tstar], __float_as_uint(tms));
    atomicMax((unsigned int*)&maxiou[b * T + tstar], __float_as_uint(iouv));
    mark = tms;
  }
  uidx[(size_t)b * A + a] = tstar;
  tmstar[(size_t)b * A + a] = mark;
}

// ---------------- k4: normalize term + cls_norm / iou-loss / x*w sums --------
__global__ void k4_losses(const float* __restrict__ target,
                          const float* __restrict__ pcls,
                          const float* __restrict__ pbox,
                          const float* __restrict__ maxtm,
                          const float* __restrict__ maxiou,
                          const int* __restrict__ uidx,
                          const float* __restrict__ tmstar,
                          float* __restrict__ accum,
                          int T, int A, int C) {
  __shared__ float sx1[T_CAP], sy1[T_CAP], sx2[T_CAP], sy2[T_CAP], sat[T_CAP];
  __shared__ int stc[T_CAP];
  int b = blockIdx.y;
  int tid = threadIdx.x;
  if (tid < T) {
    const float* tr = target + ((size_t)b * T + tid) * 5;
    int tc = (int)tr[0]; tc = tc < 0 ? 0 : (tc >= C ? C - 1 : tc);
    stc[tid] = tc;
    sx1[tid] = tr[1]; sy1[tid] = tr[2]; sx2[tid] = tr[3]; sy2[tid] = tr[4];
    sat[tid] = atanf((tr[3] - tr[1]) / (tr[4] - tr[2] + EPSF));
  }
  __syncthreads();
  int a = blockIdx.x * blockDim.x + tid;
  float cnorm = 0.0f, ciouL = 0.0f, xw = 0.0f;
  if (a < A) {
    float tms = tmstar[(size_t)b * A + a];
    if (tms >= 0.0f) {
      int t = uidx[(size_t)b * A + a];
      float w = tms / (maxtm[b * T + t] + 1e-9f) * maxiou[b * T + t];
      cnorm = w;
      float4 p = ((const float4*)(pbox + (size_t)b * A * 4))[a];
      float iou_e = ciou_eval(sx1[t], sy1[t], sx2[t], sy2[t], sat[t],
                              p.x, p.y, p.z, p.w);  // unclipped for the loss
      ciouL = (1.0f - iou_e) * w;
      float x = pcls[((size_t)b * A + a) * C + stc[t]];
      xw = x * w;
    }
  }
  float r0 = blockReduceSum(cnorm);
  float r1 = blockReduceSum(ciouL);
  float r2 = blockReduceSum(xw);
  if (tid == 0) {
    atomicAdd(&accum[0], r0);
    atomicAdd(&accum[1], r1);
    atomicAdd(&accum[2], r2);
  }
}

// ---------------- k5: sum softplus(predict_cls) over all elements ------------
__global__ void k5_softplus(const float4* __restrict__ x4, long long n4,
                            const float* __restrict__ xs, long long nrem,
                            float* __restrict__ accum) {
  long long i = (long long)blockIdx.x * blockDim.x + threadIdx.x;
  long long stride = (long long)gridDim.x * blockDim.x;
  float s = 0.0f;
  for (; i < n4; i += stride) {
    __builtin_prefetch(&x4[i + stride], 0, 0);
    float4 v = x4[i];
    s += fmaxf(v.x, 0.0f) + log1pf(expf(-fabsf(v.x)));
    s += fmaxf(v.y, 0.0f) + log1pf(expf(-fabsf(v.y)));
    s += fmaxf(v.z, 0.0f) + log1pf(expf(-fabsf(v.z)));
    s += fmaxf(v.w, 0.0f) + log1pf(expf(-fabsf(v.w)));
  }
  if (blockIdx.x == 0 && threadIdx.x == 0) {
    for (long long r = 0; r < nrem; ++r) {
      float x = xs[n4 * 4 + r];
      s += fmaxf(x, 0.0f) + log1pf(expf(-fabsf(x)));
    }
  }
  float r = blockReduceSum(s);
  if (threadIdx.x == 0) atomicAdd(&accum[3], r);
}

// ---------------- k6: finalize ------------------------------------------------
__global__ void k_final(const float* __restrict__ accum, float* __restrict__ out) {
  float cls_norm = fmaxf(accum[0], 1.0f);
  out[0] = accum[1] / cls_norm;               // loss_iou
  out[1] = (accum[3] - accum[2]) / cls_norm;  // loss_cls = sum softplus - sum x*t
}

// =============================================================================
extern "C" void kernel_launch(void* const* d_in, const int* in_sizes, int n_in,
                              void* d_out, int out_size, void* d_ws, size_t ws_size,
                              hipStream_t stream) {
  const float* target  = (const float*)d_in[0];
  const float* pcls    = (const float*)d_in[1];
  const float* pbox    = (const float*)d_in[2];
  const float* anchors = (const float*)d_in[3];

  int A = in_sizes[3] / 2;                       // 8400
  int C = 80;
  int B = (int)(in_sizes[1] / ((long long)A * C));  // 32
  int T = in_sizes[0] / (B * 5);                 // 64

  float* ws = (float*)d_ws;
  size_t o = 0;
  float* tm   = ws + o; o += (size_t)B * T * A;  // ~69MB target_matrix
  float* thr  = ws + o; o += (size_t)B * T;
  float* mtm  = ws + o; o += (size_t)B * T;
  float* mio  = ws + o; o += (size_t)B * T;
  float* tmst = ws + o; o += (size_t)B * A;
  int*   uix  = (int*)(ws + o); o += (size_t)B * A;
  float* acc  = ws + o; o += 8;

  int nbt = B * T;
  k_init<<<(nbt + BLK - 1) / BLK, BLK, 0, stream>>>(mtm, mio, acc, nbt);
  k1_target_matrix<<<nbt, BLK, 0, stream>>>(target, pcls, pbox, anchors, tm, T, A, C);
  k2_topk<<<nbt, BLK, 0, stream>>>(tm, thr, A);
  dim3 g3((A + BLK - 1) / BLK, B);
  k3_dedup<<<g3, BLK, 0, stream>>>(target, pbox, tm, thr, mtm, mio, uix, tmst, T, A);
  k4_losses<<<g3, BLK, 0, stream>>>(target, pcls, pbox, mtm, mio, uix, tmst, acc, T, A, C);
  long long N = (long long)B * A * C;
  long long n4 = N / 4;
  long long nrem = N - n4 * 4;
  k5_softplus<<<4096, BLK, 0, stream>>>((const float4*)pcls, n4, pcls, nrem, acc);
  k_final<<<1, 1, 0, stream>>>(acc, (float*)d_out);
}